// TemporalAttention_85856396248046
// MI455X (gfx1250) — compile-verified
//
#include <hip/hip_runtime.h>
#include <math.h>

typedef __attribute__((ext_vector_type(2))) float v2f;
typedef __attribute__((ext_vector_type(8))) float v8f;

#define B_      2
#define T_      8
#define NQ_     128
#define NK_     128
#define DIM_IN  64
#define DIM_OUT 256

// Hardware tanh (V_TANH_F32, single TRANS32 op on gfx1250) if available,
// else fall back to libm expansion.
#if __has_builtin(__builtin_amdgcn_tanhf)
__device__ __forceinline__ float fast_tanh(float x) { return __builtin_amdgcn_tanhf(x); }
#else
__device__ __forceinline__ float fast_tanh(float x) { return tanhf(x); }
#endif

// ---------------------------------------------------------------------------
// Projection GEMM: C[M,256] = A[M,64] * W[64,256] (+ optional bias)
// One wave32 per 16x16 output tile, native fp32 WMMA (V_WMMA_F32_16X16X4_F32).
// A 16x4 layout (ISA 7.12.2): lane m = lane&15; VGPR j holds A[m][kb+j],
//   kb = 2*(lane>>4).  B 4x16: VGPR j lane L holds W[kb+j][L&15].
// C/D 16x16: VGPR r, lanes 0-15 -> M=r, lanes 16-31 -> M=r+8, N = lane&15.
// ---------------------------------------------------------------------------
__global__ __launch_bounds__(32) void proj_wmma_kernel(const float* __restrict__ A,
                                                       const float* __restrict__ W,
                                                       const float* __restrict__ bias,
                                                       float* __restrict__ C) {
  const int lane  = threadIdx.x;            // 0..31 (wave32, EXEC all 1s)
  const int tileM = blockIdx.x;
  const int tileN = blockIdx.y;
  const int mrow  = lane & 15;
  const int kb    = (lane >> 4) << 1;       // 0 or 2
  const int n     = tileN * 16 + (lane & 15);

  const float* __restrict__ arow = A + (size_t)(tileM * 16 + mrow) * DIM_IN;

  v8f acc = {};
#pragma unroll
  for (int k0 = 0; k0 < DIM_IN; k0 += 4) {
    v2f a, b;
    a.x = arow[k0 + kb + 0];                       // contiguous pair
    a.y = arow[k0 + kb + 1];
    b.x = W[(size_t)(k0 + kb + 0) * DIM_OUT + n];
    b.y = W[(size_t)(k0 + kb + 1) * DIM_OUT + n];
    // 8 args: (neg_a, A, neg_b, B, c_mod, C, reuse_a, reuse_b)
    acc = __builtin_amdgcn_wmma_f32_16x16x4_f32(false, a, false, b,
                                                (short)0, acc, false, false);
  }

  const float bv = bias ? bias[n] : 0.0f;
#pragma unroll
  for (int r = 0; r < 8; ++r) {
    const int row = tileM * 16 + r + ((lane >> 4) << 3);
    C[(size_t)row * DIM_OUT + n] = acc[r] + bv;
  }
}

// ---------------------------------------------------------------------------
// Per (b,t,qi): scores[ki] = sum_d v[d]*tanh(q[d]+k[ki][d]); softmax over ki.
// Block = (32,8) = 8 wave32; each wave owns one ki per iteration, lane sums
// 8 strided d's (v_tanh_f32 + v_fma per element), then a 32-lane xor-shuffle
// reduction (explicit wave32 width).
// ---------------------------------------------------------------------------
__global__ __launch_bounds__(256) void score_softmax_kernel(const float* __restrict__ qp,
                                                            const float* __restrict__ kp,
                                                            const float* __restrict__ v,
                                                            float* __restrict__ out) {
  __shared__ float sq[DIM_OUT];
  __shared__ float sv[DIM_OUT];
  __shared__ float ssc[NK_];
  __shared__ float sred[NK_];

  const int qi   = blockIdx.x;
  const int t    = blockIdx.y;
  const int b    = blockIdx.z;
  const int lane = threadIdx.x;             // 0..31
  const int yy   = threadIdx.y;             // 0..7
  const int tid  = yy * 32 + lane;          // 0..255

  const float* __restrict__ qrow = qp + (size_t)(b * NQ_ + qi) * DIM_OUT;
  sq[tid] = qrow[tid];
  sv[tid] = v[tid];
  __syncthreads();

  const float* __restrict__ kbase = kp + (size_t)((b * T_ + t) * NK_) * DIM_OUT;

  for (int ki0 = 0; ki0 < NK_; ki0 += 8) {
    const int ki = ki0 + yy;
    const float* __restrict__ krow = kbase + (size_t)ki * DIM_OUT;
    float p = 0.0f;
#pragma unroll
    for (int j = 0; j < 8; ++j) {
      const int d = lane + 32 * j;          // 128B coalesced per wave
      p += sv[d] * fast_tanh(sq[d] + krow[d]);
    }
#pragma unroll
    for (int off = 16; off > 0; off >>= 1)
      p += __shfl_xor(p, off, 32);          // explicit wave32 width
    if (lane == 0) ssc[ki] = p;
  }
  __syncthreads();

  // --- softmax over 128 scores ---
  if (tid < NK_) sred[tid] = ssc[tid];
  __syncthreads();
  for (int s = 64; s > 0; s >>= 1) {
    if (tid < s) sred[tid] = fmaxf(sred[tid], sred[tid + s]);
    __syncthreads();
  }
  const float mx = sred[0];
  __syncthreads();
  float e = 0.0f;
  if (tid < NK_) { e = __expf(ssc[tid] - mx); sred[tid] = e; }
  __syncthreads();
  for (int s = 64; s > 0; s >>= 1) {
    if (tid < s) sred[tid] += sred[tid + s];
    __syncthreads();
  }
  const float inv = 1.0f / sred[0];
  if (tid < NK_)
    out[(size_t)((b * T_ + t) * NQ_ + qi) * NK_ + tid] = e * inv;
}

// ---------------------------------------------------------------------------
extern "C" void kernel_launch(void* const* d_in, const int* in_sizes, int n_in,
                              void* d_out, int out_size, void* d_ws, size_t ws_size,
                              hipStream_t stream) {
  const float* query = (const float*)d_in[0]; // [B,NQ,64]
  const float* keys  = (const float*)d_in[1]; // [B,T,NK,64]
  const float* Wq    = (const float*)d_in[2]; // [64,256]
  const float* Wk    = (const float*)d_in[3]; // [64,256]
  const float* bk    = (const float*)d_in[4]; // [256]
  const float* v     = (const float*)d_in[5]; // [256]
  float*       out   = (float*)d_out;         // [B,T,NQ,NK]

  float* qp = (float*)d_ws;                               // 2*128*256 f32
  float* kp = qp + (size_t)B_ * NQ_ * DIM_OUT;            // 2*8*128*256 f32

  // q projection: M = B*NQ = 256 -> 16 M-tiles x 16 N-tiles
  proj_wmma_kernel<<<dim3((B_ * NQ_) / 16, DIM_OUT / 16), dim3(32, 1, 1), 0, stream>>>(
      query, Wq, nullptr, qp);

  // k projection: M = B*T*NK = 2048 -> 128 M-tiles x 16 N-tiles, bias fused
  proj_wmma_kernel<<<dim3((B_ * T_ * NK_) / 16, DIM_OUT / 16), dim3(32, 1, 1), 0, stream>>>(
      keys, Wk, bk, kp);

  // fused tanh-score + softmax: one block per (b,t,qi)
  score_softmax_kernel<<<dim3(NQ_, T_, B_), dim3(32, 8, 1), 0, stream>>>(qp, kp, v, out);
}